// SlowLinear_64166811402388
// MI455X (gfx1250) — compile-verified
//
#include <hip/hip_runtime.h>

// irreps layout: 256x0e + 128x1o + 64x2e  -> row width 960 f32
// y[z,w,i] = (1/sqrt(mul)) * sum_u x[z,u,i] * W[u,w]
//
// Strategy (MI455X / gfx1250, wave32):
//  * native fp32 WMMA (v_wmma_f32_16x16x4_f32): full precision, matrix-pipe rate
//  * per-WG: stage 32 feature rows in LDS, de-interleaved to c-major layout
//    xs[(z*D+i)][u] (stride % 64 banks == 4 -> conflict-free b64 fragment reads)
//  * weights pre-packed once per launch into d_ws as K-pair float2 so each
//    lane's B fragment is ONE global_load_b64 (L2/WGP$-resident, 336 KB total)
//  * per-wave: own a 16-wide w-tile, register-block 2 c-tiles per B fragment

typedef __attribute__((ext_vector_type(2))) float v2f;
typedef __attribute__((ext_vector_type(8))) float v8f;

#define ROWW 960  // full feature row width (f32 elements)

// Pack W[mul][mul] -> Wp[k/2][w][2] = {W[2p][w], W[2p+1][w]}
__global__ void pack_weights_kernel(const float* __restrict__ w,
                                    float* __restrict__ wp, int mul) {
  const int idx = blockIdx.x * blockDim.x + threadIdx.x;
  const int total = (mul >> 1) * mul;
  if (idx < total) {
    const int p = idx / mul;
    const int col = idx - p * mul;
    wp[2 * idx + 0] = w[(2 * p) * mul + col];
    wp[2 * idx + 1] = w[(2 * p + 1) * mul + col];
  }
}

template <int MUL, int D, int OFF, int ZT, int NWAVES, bool PACKED>
__global__ __launch_bounds__(NWAVES * 32) void irreps_linear_kernel(
    const float* __restrict__ feat, const float* __restrict__ wsrc,
    float* __restrict__ out, float scale) {
  constexpr int W    = MUL * D;   // floats of this irrep block per row
  constexpr int F4   = W / 4;     // float4 per feature row
  constexpr int MULp = MUL + 4;   // padded LDS stride (MULp % 64 == 4)
  constexpr int WT   = MUL / 16;  // output-channel tiles

  __shared__ float xs[ZT * D * MULp];  // c-major: xs[c = z*D + i][u]

  const int tid  = threadIdx.x;
  const int lane = tid & 31;
  const int wave = tid >> 5;
  const int zg0  = blockIdx.x * ZT;

  // ---- stage feature tile: coalesced float4 global reads, de-interleave
  //      m-components into c-major LDS layout ----
  for (int idx = tid; idx < ZT * F4; idx += NWAVES * 32) {
    const int r  = idx / F4;
    const int c4 = idx - r * F4;
    const float4 v =
        *(const float4*)(feat + (size_t)(zg0 + r) * ROWW + OFF + c4 * 4);
    const float vv[4] = {v.x, v.y, v.z, v.w};
#pragma unroll
    for (int j = 0; j < 4; ++j) {
      const int t = c4 * 4 + j;          // u*D + i
      xs[(r * D + t % D) * MULp + t / D] = vv[j];
    }
  }
  __syncthreads();

  const int mrow  = lane & 15;        // M row (A) / N col (B,D) within tile
  const int khalf = (lane >> 4) * 2;  // K base for this half-wave
  const int mhi   = (lane >> 4) * 8;  // D-matrix M offset for upper lanes

  for (int wt = wave; wt < WT; wt += NWAVES) {  // wave-uniform loop
    const int w0 = wt * 16;
    // per-lane packed-B base: element ((k/2 + lane>>4)*MUL + w0+mrow)*2
    const float* bp =
        PACKED ? wsrc + 2 * ((size_t)(lane >> 4) * MUL + w0 + mrow) : nullptr;
    for (int pass = 0; pass < D; ++pass) {
      const int c0 = pass * 32;  // two 16-row c-tiles: [c0, c0+16) and [c0+16, c0+32)
      const float* a0p = xs + (size_t)(c0 + mrow) * MULp;
      const float* a1p = a0p + 16 * MULp;
      v8f acc0 = {};
      v8f acc1 = {};
      for (int k = 0; k < MUL; k += 4) {
        const int kb = k + khalf;
        v2f b;
        if constexpr (PACKED) {
          b = *(const v2f*)(bp + (size_t)k * MUL);  // (k/2)*2*MUL == k*MUL
        } else {
          b.x = wsrc[(size_t)kb * MUL + w0 + mrow];
          b.y = wsrc[(size_t)(kb + 1) * MUL + w0 + mrow];
        }
        const v2f a0 = *(const v2f*)(a0p + kb);  // Xv[c, kb..kb+1]
        const v2f a1 = *(const v2f*)(a1p + kb);
        acc0 = __builtin_amdgcn_wmma_f32_16x16x4_f32(
            false, a0, false, b, (short)0, acc0, false, false);
        acc1 = __builtin_amdgcn_wmma_f32_16x16x4_f32(
            false, a1, false, b, (short)0, acc1, false, false);
      }
      // D layout: vgpr r, lanes0-15 -> M=r, lanes16-31 -> M=r+8; N = mrow
#pragma unroll
      for (int r = 0; r < 8; ++r) {
        const int cm0 = c0 + r + mhi;
        const int cm1 = cm0 + 16;
        out[(size_t)(zg0 + cm0 / D) * ROWW + OFF + (w0 + mrow) * D +
            (cm0 % D)] = acc0[r] * scale;
        out[(size_t)(zg0 + cm1 / D) * ROWW + OFF + (w0 + mrow) * D +
            (cm1 % D)] = acc1[r] * scale;
      }
    }
  }
}

extern "C" void kernel_launch(void* const* d_in, const int* in_sizes, int n_in,
                              void* d_out, int out_size, void* d_ws,
                              size_t ws_size, hipStream_t stream) {
  (void)n_in; (void)out_size;
  const float* feat = (const float*)d_in[0];
  const float* w0   = (const float*)d_in[1];
  const float* w1   = (const float*)d_in[2];
  const float* w2   = (const float*)d_in[3];
  float* out = (float*)d_out;
  float* ws  = (float*)d_ws;

  const int N = in_sizes[0] / ROWW;  // 131072
  constexpr int ZT = 32;
  dim3 block(128);  // 4 waves
  dim3 grid(N / ZT);

  const float s0 = 0.0625f;               // 1/sqrt(256)
  const float s1 = 0.08838834764831845f;  // 1/sqrt(128)
  const float s2 = 0.125f;                // 1/sqrt(64)

  // packed weight buffers in workspace: 256KB + 64KB + 16KB = 336KB
  const bool packed = ws_size >= (size_t)(65536 + 16384 + 4096) * 4;
  if (packed) {
    float* wp0 = ws;
    float* wp1 = ws + 65536;
    float* wp2 = ws + 65536 + 16384;
    pack_weights_kernel<<<dim3(128), dim3(256), 0, stream>>>(w0, wp0, 256);
    pack_weights_kernel<<<dim3(32), dim3(256), 0, stream>>>(w1, wp1, 128);
    pack_weights_kernel<<<dim3(8), dim3(256), 0, stream>>>(w2, wp2, 64);
    irreps_linear_kernel<256, 1, 0, ZT, 4, true>
        <<<grid, block, 0, stream>>>(feat, wp0, out, s0);
    irreps_linear_kernel<128, 3, 256, ZT, 4, true>
        <<<grid, block, 0, stream>>>(feat, wp1, out, s1);
    irreps_linear_kernel<64, 5, 640, ZT, 4, true>
        <<<grid, block, 0, stream>>>(feat, wp2, out, s2);
  } else {
    irreps_linear_kernel<256, 1, 0, ZT, 4, false>
        <<<grid, block, 0, stream>>>(feat, w0, out, s0);
    irreps_linear_kernel<128, 3, 256, ZT, 4, false>
        <<<grid, block, 0, stream>>>(feat, w1, out, s1);
    irreps_linear_kernel<64, 5, 640, ZT, 4, false>
        <<<grid, block, 0, stream>>>(feat, w2, out, s2);
  }
}